// GCBlock_21723944583554
// MI455X (gfx1250) — compile-verified
//
#include <hip/hip_runtime.h>
#include <stdint.h>

#define V_DIM 66
#define T_DIM 256
#define J_DIM 22
#define MPAD  80          // V padded to multiple of 16 for WMMA M-tiles
#define SROW  258         // LDS row stride (floats) for x / pre  -> conflict-free A loads
#define FROW  257         // LDS row stride (floats) for fc_w chunk
#define KC    64          // K chunk staged in LDS

typedef __attribute__((ext_vector_type(2))) float v2f;
typedef __attribute__((ext_vector_type(8))) float v8f;

// ---------------------------------------------------------------------------
// Exact JAX threefry2x32 (key = (0,42)) so the gumbel argmax gate matches.
// ---------------------------------------------------------------------------
__device__ __forceinline__ uint32_t rotl32(uint32_t x, int r) {
  return (x << r) | (x >> (32 - r));
}

__device__ __forceinline__ void threefry2x32(uint32_t k0, uint32_t k1,
                                             uint32_t x0, uint32_t x1,
                                             uint32_t& o0, uint32_t& o1) {
  uint32_t ks[3] = {k0, k1, k0 ^ k1 ^ 0x1BD11BDAu};
  x0 += ks[0]; x1 += ks[1];
  const int rot0[4] = {13, 15, 26, 6};
  const int rot1[4] = {17, 29, 16, 24};
  #pragma unroll
  for (int g = 0; g < 5; ++g) {
    #pragma unroll
    for (int i = 0; i < 4; ++i) {
      int r = (g & 1) ? rot1[i] : rot0[i];
      x0 += x1; x1 = rotl32(x1, r); x1 ^= x0;
    }
    x0 += ks[(g + 1) % 3];
    x1 += ks[(g + 2) % 3] + (uint32_t)(g + 1);
  }
  o0 = x0; o1 = x1;
}

// ---------------------------------------------------------------------------
// Kernel 1: per-batch gate selection.
// prob[b,k] = (mean_v x[b,v,:]) @ mlp[:,k]; gate = argmax(prob + gumbel)
// (straight-through estimator's forward value is exactly the one-hot argmax).
// ---------------------------------------------------------------------------
__global__ void __launch_bounds__(256)
gc_gate_kernel(const float* __restrict__ x, const float* __restrict__ mlp,
               const int* __restrict__ dyn, int* __restrict__ gate, int B) {
  __shared__ float red[256];
  __shared__ float probs[4];
  const int b = blockIdx.x, tid = threadIdx.x;
  const float* xb = x + (size_t)b * V_DIM * T_DIM;

  float s = 0.f;
  #pragma unroll 6
  for (int v = 0; v < V_DIM; ++v) s += xb[v * T_DIM + tid];
  const float xm = s * (1.0f / (float)V_DIM);

  for (int k = 0; k < 4; ++k) {
    red[tid] = xm * mlp[tid * 4 + k];
    __syncthreads();
    for (int off = 128; off > 0; off >>= 1) {
      if (tid < off) red[tid] += red[tid + off];
      __syncthreads();
    }
    if (tid == 0) probs[k] = red[0];
    __syncthreads();
  }

  if (tid == 0) {
    int gi = 0;
    if (dyn[0]) {
      const uint32_t half = (uint32_t)(B * 2);  // total draws = B*4, split in half
      float best = -__builtin_inff();
      for (int k = 0; k < 4; ++k) {
        uint32_t idx = (uint32_t)(b * 4 + k);
        uint32_t i0 = (idx < half) ? idx : (idx - half);
        uint32_t o0, o1;
        threefry2x32(0u, 42u, i0, i0 + half, o0, o1);
        uint32_t bits = (idx < half) ? o0 : o1;
        uint32_t mant = (bits >> 9) | 0x3f800000u;
        float f; __builtin_memcpy(&f, &mant, 4);
        float u = (f - 1.0f) * (1.0f - 1e-10f) + 1e-10f;
        u = fmaxf(u, 1e-10f);
        float gum = -__logf(-__logf(u));
        float z = probs[k] + gum;
        if (z > best) { best = z; gi = k; }
      }
    }
    gate[b] = gi;
  }
}

// ---------------------------------------------------------------------------
// Kernel 2: fused  pre = x1 + gated_mix;  h = pre @ fc_w^T (WMMA f32 16x16x4);
//           LayerNorm over v; out = x + h_ln.  One workgroup (8 waves) per b.
// LDS: region A (17028 f): x[b] during pre-phase, fc_w K-chunks during GEMM.
//      region B (20640 f): pre (80x258), then h.
// ---------------------------------------------------------------------------
__global__ void __launch_bounds__(256)
gc_main_kernel(const float* __restrict__ x, const float* __restrict__ adj_j,
               const float* __restrict__ adj_t, const float* __restrict__ adj_jc,
               const float* __restrict__ adj_tj, const float* __restrict__ fc_w,
               const float* __restrict__ alpha, const float* __restrict__ beta,
               const int* __restrict__ gsel, float* __restrict__ out) {
  __shared__ float smem[V_DIM * SROW + MPAD * SROW];   // 150,672 bytes
  __shared__ float sadjj[J_DIM * J_DIM];
  __shared__ float sadjc[J_DIM * 9];
  __shared__ float salpha[V_DIM];
  __shared__ float sbeta[V_DIM];

  float* sxf = smem;                 // x (66x258) / fc_w chunk (64x257)
  float* sp  = smem + V_DIM * SROW;  // pre then h (80x258)

  const int b   = blockIdx.x;
  const int tid = threadIdx.x;
  const size_t base = (size_t)b * V_DIM * T_DIM;

  // ---- stage x[b], adjacencies, affine params; zero pad rows of pre ----
  #pragma unroll 6
  for (int v = 0; v < V_DIM; ++v) sxf[v * SROW + tid] = x[base + v * T_DIM + tid];
  for (int i = tid; i < J_DIM * J_DIM; i += 256) sadjj[i] = adj_j[i];
  for (int i = tid; i < J_DIM * 9; i += 256)     sadjc[i] = adj_jc[i];
  if (tid < V_DIM) { salpha[tid] = alpha[tid]; sbeta[tid] = beta[tid]; }
  #pragma unroll
  for (int v = V_DIM; v < MPAD; ++v) sp[v * SROW + tid] = 0.0f;
  __syncthreads();

  // ---- pre[v, t=tid] = x1 + gated mix ----
  const int g = gsel[b];
  float atm = 0.f, atp = 0.f;
  if (g == 1) {  // shared banded adj_t coefficients for column t=tid
    if (tid > 0)         atm = adj_t[tid * T_DIM + tid - 1];
    if (tid < T_DIM - 1) atp = adj_t[tid * T_DIM + tid + 1];
  }
  for (int v = 0; v < V_DIM; ++v) {
    const int jo = v / 3, c = v % 3;
    float s = 0.f;
    #pragma unroll
    for (int j = 0; j < J_DIM; ++j)
      s += sadjj[jo * J_DIM + j] * sxf[(3 * j + c) * SROW + tid];
    float m = 0.f;
    if (g == 1) {
      if (tid > 0)         m += atm * sxf[v * SROW + tid - 1];
      if (tid < T_DIM - 1) m += atp * sxf[v * SROW + tid + 1];
    } else if (g == 2) {
      #pragma unroll
      for (int cc = 0; cc < 3; ++cc)
        m += sadjc[jo * 9 + c * 3 + cc] * sxf[(3 * jo + cc) * SROW + tid];
    } else if (g == 3) {
      const float* at = adj_tj + (size_t)v * T_DIM * T_DIM + (size_t)tid * T_DIM;
      if (tid > 0)         m += at[tid - 1] * sxf[v * SROW + tid - 1];
      if (tid < T_DIM - 1) m += at[tid + 1] * sxf[v * SROW + tid + 1];
    }
    sp[v * SROW + tid] = s + m;
  }
  __syncthreads();

  // ---- GEMM: h[v,f] = sum_t pre[v,t] * fc_w[f,t] via V_WMMA_F32_16X16X4_F32
  const int wave  = tid >> 5;           // 0..7  -> N columns {2w,2w+1}*16
  const int lane  = tid & 31;
  const int lrow  = lane & 15;          // M row / N col within tile
  const int khalf = lane >> 4;          // K half-split (0: k0,k0+1  1: k0+2,k0+3)

  v8f acc[5][2];
  #pragma unroll
  for (int mt = 0; mt < 5; ++mt)
    #pragma unroll
    for (int nt = 0; nt < 2; ++nt) acc[mt][nt] = (v8f)0.0f;

  for (int kk = 0; kk < T_DIM / KC; ++kk) {
    // stage transposed fc_w chunk: sxf[k*FROW + f] = fc_w[f, kk*KC + k]
    #pragma unroll 8
    for (int i = 0; i < 64; ++i) {
      int j = i * 256 + tid;
      int f = j >> 6, k = j & 63;
      sxf[k * FROW + f] = fc_w[f * T_DIM + kk * KC + k];
    }
    __syncthreads();

    for (int k4 = 0; k4 < KC / 4; ++k4) {
      const int kl = k4 * 4 + 2 * khalf;   // this lane's K pair within chunk
      v2f af[5];
      #pragma unroll
      for (int mt = 0; mt < 5; ++mt)
        af[mt] = *(const v2f*)&sp[(mt * 16 + lrow) * SROW + kk * KC + kl];
      v2f bf[2];
      #pragma unroll
      for (int nt = 0; nt < 2; ++nt) {
        const int n0 = (wave * 2 + nt) * 16;
        bf[nt].x = sxf[kl * FROW + n0 + lrow];
        bf[nt].y = sxf[(kl + 1) * FROW + n0 + lrow];
      }
      #pragma unroll
      for (int mt = 0; mt < 5; ++mt)
        #pragma unroll
        for (int nt = 0; nt < 2; ++nt)
          acc[mt][nt] = __builtin_amdgcn_wmma_f32_16x16x4_f32(
              false, af[mt], false, bf[nt], (short)0, acc[mt][nt], false, false);
    }
    __syncthreads();
  }

  // ---- write h tiles back to LDS (overwriting pre) ----
  #pragma unroll
  for (int mt = 0; mt < 5; ++mt)
    #pragma unroll
    for (int nt = 0; nt < 2; ++nt) {
      const int col = (wave * 2 + nt) * 16 + lrow;
      #pragma unroll
      for (int r = 0; r < 8; ++r) {
        const int row = mt * 16 + r + 8 * khalf;
        sp[row * SROW + col] = acc[mt][nt][r];
      }
    }
  __syncthreads();

  // ---- LayerNorm over v (fc_b is constant over v => cancels exactly) ----
  float sum = 0.f, sq = 0.f;
  #pragma unroll 6
  for (int v = 0; v < V_DIM; ++v) {
    float h = sp[v * SROW + tid];
    sum += h; sq += h * h;
  }
  const float mean = sum * (1.0f / (float)V_DIM);
  const float var  = sq * (1.0f / (float)V_DIM) - mean * mean;
  const float rstd = rsqrtf(var + 1e-5f);

  #pragma unroll 6
  for (int v = 0; v < V_DIM; ++v) {
    float h  = sp[v * SROW + tid];
    float xv = x[base + v * T_DIM + tid];   // L2-hot re-read (sxf was reused)
    out[base + v * T_DIM + tid] =
        xv + (h - mean) * rstd * salpha[v] + sbeta[v];
  }
}

// ---------------------------------------------------------------------------
extern "C" void kernel_launch(void* const* d_in, const int* in_sizes, int n_in,
                              void* d_out, int out_size, void* d_ws, size_t ws_size,
                              hipStream_t stream) {
  const float* x      = (const float*)d_in[0];
  const float* mlp    = (const float*)d_in[1];
  const float* adj_j  = (const float*)d_in[2];
  const float* adj_t  = (const float*)d_in[3];
  const float* adj_jc = (const float*)d_in[4];
  const float* adj_tj = (const float*)d_in[5];
  const float* fc_w   = (const float*)d_in[6];
  // d_in[7] = fc_b: constant over the LN axis -> cancels exactly, unused.
  const float* alpha  = (const float*)d_in[8];
  const float* beta   = (const float*)d_in[9];
  const int*   dyn    = (const int*)d_in[10];

  const int B = in_sizes[0] / (V_DIM * T_DIM);
  int* gate = (int*)d_ws;

  gc_gate_kernel<<<B, 256, 0, stream>>>(x, mlp, dyn, gate, B);
  gc_main_kernel<<<B, 256, 0, stream>>>(x, adj_j, adj_t, adj_jc, adj_tj,
                                        fc_w, alpha, beta, gate, (float*)d_out);
}